// SparseAttentionHead_76012331205261
// MI455X (gfx1250) — compile-verified
//
#include <hip/hip_runtime.h>

typedef __attribute__((ext_vector_type(2))) float v2f;
typedef __attribute__((ext_vector_type(4))) float v4f;
typedef __attribute__((ext_vector_type(8))) float v8f;

#define T_SEQ 4096
#define BATCH 4
#define CDIM  64
#define HDIM  64
#define NROWS (BATCH * T_SEQ)   // 16384

// ---------------------------------------------------------------------------
// Kernel 1: zero-fill the dense attn output with 128-bit non-temporal stores.
// 256 MB of zeros -> keep it out of L2 (won't fit: 256MB > 192MB) and let the
// stores stream straight to HBM. This is >90% of total bytes moved.
// ---------------------------------------------------------------------------
__global__ __launch_bounds__(256) void zero_fill_nt(v4f* __restrict__ p, long n4) {
    long i = (long)blockIdx.x * blockDim.x + threadIdx.x;
    long stride = (long)gridDim.x * blockDim.x;
    v4f z = {0.f, 0.f, 0.f, 0.f};
    for (; i < n4; i += stride)
        __builtin_nontemporal_store(z, p + i);
}

// ---------------------------------------------------------------------------
// Kernel 2: q/k/v = x @ W^T via V_WMMA_F32_16X16X4_F32.
// One wave computes one 16x16 output tile; K=64 swept in 16 steps of 4.
// ISA layout for 32-bit A (16x4) / B (4x16):
//   lane = r + 16*half;  A: a[j] = A[r][2*half + j]   (row r, K-pair)
//                        B: b[j] = B[2*half + j][r]   (col r, K-pair)
//   C/D: vgpr i, lane -> D[i + 8*half][r]
// With y = x @ W^T:  A = x tile,  B[k][n] = W[n][k] -> both operands are
// contiguous float pairs per lane (coalesced 8-byte loads).
// ---------------------------------------------------------------------------
__global__ __launch_bounds__(256) void qkv_wmma(
    const float* __restrict__ x,
    const float* __restrict__ Wq,
    const float* __restrict__ Wk,
    const float* __restrict__ Wv,
    float* __restrict__ q, float* __restrict__ k, float* __restrict__ v) {

    int wave = (int)((blockIdx.x * blockDim.x + threadIdx.x) >> 5);
    int lane = threadIdx.x & 31;

    int mtile = wave & 1023;       // 1024 tiles of 16 rows (16384 rows)
    int rest  = wave >> 10;
    int ntile = rest & 3;          // 4 tiles of 16 cols (64 cols)
    int mat   = rest >> 2;         // 0=q, 1=k, 2=v

    const float* W = (mat == 0) ? Wq : (mat == 1) ? Wk : Wv;
    float*       Y = (mat == 0) ? q  : (mat == 1) ? k  : v;

    int half = lane >> 4;          // K-pair selector
    int r    = lane & 15;          // M index (A) / N index (B)
    int m0   = mtile * 16;
    int n0   = ntile * 16;

    const float* xrow = x + (long)(m0 + r) * CDIM;   // A row (M = m0 + r)
    const float* wrow = W + (long)(n0 + r) * CDIM;   // W row n -> B column n

    v8f acc = {};
#pragma unroll
    for (int kk = 0; kk < CDIM; kk += 4) {
        v2f a, b;
        a.x = xrow[kk + 2 * half + 0];
        a.y = xrow[kk + 2 * half + 1];
        b.x = wrow[kk + 2 * half + 0];
        b.y = wrow[kk + 2 * half + 1];
        acc = __builtin_amdgcn_wmma_f32_16x16x4_f32(
            /*neg_a=*/false, a, /*neg_b=*/false, b,
            /*c_mod=*/(short)0, acc, /*reuse_a=*/false, /*reuse_b=*/false);
    }

#pragma unroll
    for (int i = 0; i < 8; ++i) {
        int m = m0 + i + 8 * half;
        Y[(long)m * HDIM + n0 + r] = acc[i];
    }
}

// ---------------------------------------------------------------------------
// Kernel 3: banded softmax + op, one thread per (b,t) row.
// Band: s in [t-2, t+2] clipped. Bias +1.0 for s>t (triu quirk of reference).
// Writes the <=5 nonzero attn entries (on top of the NT zero-fill) and the
// 64-wide op row with float4 traffic.
// ---------------------------------------------------------------------------
__global__ __launch_bounds__(256) void band_attn(
    const float* __restrict__ q,
    const float* __restrict__ k,
    const float* __restrict__ v,
    float* __restrict__ op,      // [B*T, 64]
    float* __restrict__ attn) {  // [B, T, T]

    int row = blockIdx.x * blockDim.x + threadIdx.x;  // b*T + t
    if (row >= NROWS) return;
    int b = row >> 12;           // /4096
    int t = row & (T_SEQ - 1);

    int s_lo = (t - 2 < 0) ? 0 : t - 2;
    int s_hi = (t + 2 > T_SEQ - 1) ? T_SEQ - 1 : t + 2;
    int n = s_hi - s_lo + 1;

    const float4* qv = (const float4*)(q + (long)row * HDIM);
    float4 qr[16];
#pragma unroll
    for (int j = 0; j < 16; ++j) qr[j] = qv[j];

    float sc[5];
    float mmax = -1e30f;
    for (int i = 0; i < n; ++i) {
        int s = s_lo + i;
        const float4* kv = (const float4*)(k + ((long)b * T_SEQ + s) * HDIM);
        float d = 0.f;
#pragma unroll
        for (int j = 0; j < 16; ++j) {
            float4 a = qr[j], c = kv[j];
            d += a.x * c.x + a.y * c.y + a.z * c.z + a.w * c.w;
        }
        float val = d * 0.125f + ((s > t) ? 1.0f : 0.0f);  // scale = 64^-0.5
        sc[i] = val;
        mmax = fmaxf(mmax, val);
    }

    float denom = 0.f;
    for (int i = 0; i < n; ++i) { sc[i] = __expf(sc[i] - mmax); denom += sc[i]; }
    float inv = 1.0f / denom;

    float* arow = attn + (long)b * T_SEQ * T_SEQ + (long)t * T_SEQ;
    for (int i = 0; i < n; ++i) {
        sc[i] *= inv;
        arow[s_lo + i] = sc[i];
    }

    float4 o[16];
#pragma unroll
    for (int j = 0; j < 16; ++j) o[j] = make_float4(0.f, 0.f, 0.f, 0.f);
    for (int i = 0; i < n; ++i) {
        const float4* vv = (const float4*)(v + ((long)b * T_SEQ + s_lo + i) * HDIM);
        float w = sc[i];
#pragma unroll
        for (int j = 0; j < 16; ++j) {
            float4 c = vv[j];
            o[j].x += w * c.x; o[j].y += w * c.y;
            o[j].z += w * c.z; o[j].w += w * c.w;
        }
    }
    float4* orow = (float4*)(op + (long)row * HDIM);
#pragma unroll
    for (int j = 0; j < 16; ++j) orow[j] = o[j];
}

// ---------------------------------------------------------------------------
extern "C" void kernel_launch(void* const* d_in, const int* in_sizes, int n_in,
                              void* d_out, int out_size, void* d_ws, size_t ws_size,
                              hipStream_t stream) {
    const float* x  = (const float*)d_in[0];
    const float* Wq = (const float*)d_in[1];
    const float* Wk = (const float*)d_in[2];
    const float* Wv = (const float*)d_in[3];

    float* op   = (float*)d_out;                          // B*T*64
    float* attn = op + (long)NROWS * HDIM;                // B*T*T

    float* q = (float*)d_ws;                              // 4 MB each
    float* k = q + (long)NROWS * HDIM;
    float* v = k + (long)NROWS * HDIM;

    // 1) Stream 256 MB of zeros into attn with NT stores.
    long attn_n4 = (long)BATCH * T_SEQ * T_SEQ / 4;
    zero_fill_nt<<<4096, 256, 0, stream>>>((v4f*)attn, attn_n4);

    // 2) QKV projection: 1024 m-tiles x 4 n-tiles x 3 matrices = 12288 waves,
    //    8 waves per 256-thread block -> 1536 blocks.
    qkv_wmma<<<1536, 256, 0, stream>>>(x, Wq, Wk, Wv, q, k, v);

    // 3) Banded softmax + op + band attn writes.
    band_attn<<<(NROWS + 255) / 256, 256, 0, stream>>>(q, k, v, op, attn);
}